// KnowledgeDistillationGeometricJSLoss_4707284156573
// MI455X (gfx1250) — compile-verified
//
#include <hip/hip_runtime.h>

typedef float v2f __attribute__((ext_vector_type(2)));
typedef float v4f __attribute__((ext_vector_type(4)));
typedef float v8f __attribute__((ext_vector_type(8)));

#define EPSC 1e-6f
#define REGC 1e-6f
#define LN2F 0.69314718055994530942f

// Wave32 reduction via V_WMMA_F32_16X16X4_F32:
//   A (16x4 f32): lane L<16 holds (M=L, K=0)=a.x, (M=L,K=1)=a.y
//                 lane L>=16 holds (M=L-16, K=2)=a.x, (M=L-16,K=3)=a.y
//   B = ones(4x16)  (all lanes/all VGPRs = 1.0 -> layout independent)
//   D[m][n] = sum_k A[m][k] = p[m] + p[m+16], replicated across n.
//   Summing D's 8 VGPRs: lanes 0-15 get rows 0-7 total, lanes 16-31 rows 8-15.
//   readlane(0) + readlane(16) = full 32-lane sum.
__device__ __forceinline__ float wave_sum_wmma(float p) {
    v2f a; a.x = p;    a.y = 0.0f;
    v2f b; b.x = 1.0f; b.y = 1.0f;
    v8f c = {0.f, 0.f, 0.f, 0.f, 0.f, 0.f, 0.f, 0.f};
    v8f d = __builtin_amdgcn_wmma_f32_16x16x4_f32(
        /*neg_a=*/false, a, /*neg_b=*/false, b,
        /*c_mod=*/(short)0, c, /*reuse_a=*/false, /*reuse_b=*/false);
    float t = d[0] + d[1] + d[2] + d[3] + d[4] + d[5] + d[6] + d[7];
    float lo = __builtin_bit_cast(float, __builtin_amdgcn_readlane(__builtin_bit_cast(int, t), 0));
    float hi = __builtin_bit_cast(float, __builtin_amdgcn_readlane(__builtin_bit_cast(int, t), 16));
    return lo + hi;
}

// Block (256 threads = 8 waves) reduction; thread 0 stores s*scale to *dst.
__device__ __forceinline__ void block_reduce_store(float p, float* dst, float scale) {
    __shared__ float sm[8];
    float w = wave_sum_wmma(p);           // EXEC all-1s here (pre-divergence)
    int lane = threadIdx.x & 31;
    int wid  = threadIdx.x >> 5;
    if (lane == 0) sm[wid] = w;
    __syncthreads();
    if (threadIdx.x == 0) {
        float s = 0.f;
        int nw = (blockDim.x + 31) >> 5;
        for (int i = 0; i < nw; ++i) s += sm[i];
        *dst = s * scale;
    }
}

__global__ void __launch_bounds__(256)
js_loss_partial(const v4f* __restrict__ stu, const v4f* __restrict__ tea,
                float* __restrict__ partial, int n) {
    int tid    = blockIdx.x * blockDim.x + threadIdx.x;
    int stride = gridDim.x * blockDim.x;
    float acc = 0.f;
    for (int i = tid; i < n; i += stride) {
        // Streamed once; 256MB > 192MB L2 -> non-temporal B128 loads.
        v4f s = __builtin_nontemporal_load(stu + i);   // l,t,r,b
        v4f t = __builtin_nontemporal_load(tea + i);
        // means: (log l + log r)/2 == 0.5*log(l*r)  (one v_log per axis)
        float pmx = 0.5f * __logf(fmaxf(s[0], EPSC) * fmaxf(s[2], EPSC));
        float pmy = 0.5f * __logf(fmaxf(s[1], EPSC) * fmaxf(s[3], EPSC));
        float tmx = 0.5f * __logf(fmaxf(t[0], EPSC) * fmaxf(t[2], EPSC));
        float tmy = 0.5f * __logf(fmaxf(t[1], EPSC) * fmaxf(t[3], EPSC));
        // diag covariances: center^2 + REG (a2 == center_x^2 algebraically)
        float pcx = pmx * pmx + REGC, pcy = pmy * pmy + REGC;
        float tcx = tmx * tmx + REGC, tcy = tmy * tmy + REGC;
        float Sx = pcx + tcx, Sy = pcy + tcy;     // sums
        float Px = pcx * tcx, Py = pcy * tcy;     // products
        float dx = tmx - pmx, dy = tmy - pmy;
        // trace + mahalanobis terms per axis: 0.25*(S^3 + d^2*(S^2-2P))/(P*S)
        float rx = __builtin_amdgcn_rcpf(Px * Sx);
        float ry = __builtin_amdgcn_rcpf(Py * Sy);
        float termx = 0.25f * rx * (Sx * Sx * Sx + dx * dx * (Sx * Sx - 2.f * Px));
        float termy = 0.25f * ry * (Sy * Sy * Sy + dy * dy * (Sy * Sy - 2.f * Py));
        // logdet = 2*ln2 + 0.5*log(Px*Py) - log(Sx*Sy)   (two v_log total)
        float logdet = 2.f * LN2F + 0.5f * __logf(Px * Py) - __logf(Sx * Sy);
        float js = 0.5f * (termx + termy + logdet - 2.f);
        acc += 1.f - __builtin_amdgcn_rcpf(1.f + js * js);   // TAU = 1
    }
    block_reduce_store(acc, partial + blockIdx.x, 1.0f);
}

__global__ void __launch_bounds__(256)
js_loss_final(const float* __restrict__ partial, float* __restrict__ out,
              int nb, float inv_n) {
    float acc = 0.f;
    for (int i = (int)threadIdx.x; i < nb; i += (int)blockDim.x) acc += partial[i];
    block_reduce_store(acc, out, inv_n);
}

extern "C" void kernel_launch(void* const* d_in, const int* in_sizes, int n_in,
                              void* d_out, int out_size, void* d_ws, size_t ws_size,
                              hipStream_t stream) {
    const v4f* stu = (const v4f*)d_in[0];
    const v4f* tea = (const v4f*)d_in[1];
    float* out     = (float*)d_out;
    float* partial = (float*)d_ws;
    int n = in_sizes[0] / 4;          // number of boxes
    int nb = 2048;                    // 2048 blocks * 256 thr = 16 elems/thread
    if ((size_t)nb * sizeof(float) > ws_size) nb = (int)(ws_size / sizeof(float));
    if (nb < 1) nb = 1;
    js_loss_partial<<<nb, 256, 0, stream>>>(stu, tea, partial, n);
    js_loss_final<<<1, 256, 0, stream>>>(partial, out, nb, 1.0f / (float)n);
}